// HyperbolicScaleHead_61125974556910
// MI455X (gfx1250) — compile-verified
//
#include <hip/hip_runtime.h>
#include <hip/hip_bf16.h>
#include <math.h>

// ---------------------------------------------------------------------------
// CDNA5 (gfx1250) wave32 WMMA f32 GEMM pipeline for the hyperbolic-head model.
// C = A(MxK) @ B(KxN) (+bias), epilogues: 0=bias, 1=bias+gelu, 2=hyperbolic.
// Each wave computes TN independent 16x16 tiles (independent WMMA dep chains)
// sharing one A fragment. Tail tiles are handled by CLAMPING their column base
// to N-16 (N is always a multiple of 16) so the K-loop is completely
// branch-free -> no phi/register-copy storms around the WMMAs; only the
// epilogue stores are guarded. LDB is a template param so B loads use
// immediate offsets (clauses of global_load with ioffset).
// ---------------------------------------------------------------------------

typedef __attribute__((ext_vector_type(2))) float v2f;
typedef __attribute__((ext_vector_type(8))) float v8f;

#define KSTEP 64
#define WPB 8   // waves per block (256 threads)

__device__ __forceinline__ float gelu_exact(float x) {
  return 0.5f * x * (1.0f + erff(x * 0.7071067811865475f));
}

__device__ __forceinline__ float wave_sum(float s) {
  #pragma unroll
  for (int off = 16; off > 0; off >>= 1) s += __shfl_xor(s, off, 32);
  return s;
}

__device__ __forceinline__ float curvature(const float* log_c) {
  float c = fminf(fmaxf(__expf(log_c[0]), 1e-4f), 10.0f);
  return fmaxf(fabsf(c), 1e-6f);
}

template <int EPI, int TN, int LDB>
__global__ __launch_bounds__(256)
void wmma_gemm_nn(const float* __restrict__ A, int lda,
                  const float* __restrict__ B,
                  const float* __restrict__ bias,
                  float* __restrict__ C, int ldc,
                  int M, int N, int K,
                  const float* __restrict__ rowE2,
                  const float* __restrict__ rowFe,
                  const float* __restrict__ colP2,
                  const float* __restrict__ colFp,
                  const float* __restrict__ log_c)
{
  __shared__ float As[16 * KSTEP];   // 4 KB A macro-tile shared by 8 waves

  const int lane = threadIdx.x & 31;
  const int half = lane >> 4;        // 0: lanes 0-15, 1: lanes 16-31
  const int l16  = lane & 15;
  const int wave = __builtin_amdgcn_readfirstlane(threadIdx.x >> 5);
  const int m0   = blockIdx.y * 16;
  const int n0   = __builtin_amdgcn_readfirstlane((blockIdx.x * WPB + wave) * (16 * TN));

  // Clamped per-tile column bases (scalar). N is a multiple of 16, so the
  // clamped loads are always in-bounds; tail waves just redo the last tile.
  int co[TN];
  #pragma unroll
  for (int j = 0; j < TN; ++j) {
    const int nj = n0 + 16 * j;
    co[j] = __builtin_amdgcn_readfirstlane(nj <= N - 16 ? nj : N - 16);
  }

  v8f acc[TN] = {};

  for (int k0 = 0; k0 < K; k0 += KSTEP) {
    // Cooperative, coalesced stage of the 16xKSTEP A tile into LDS.
    #pragma unroll
    for (int i = threadIdx.x; i < 16 * KSTEP; i += 256) {
      const int r = i / KSTEP;
      const int c = i % KSTEP;
      As[i] = A[(size_t)(m0 + r) * lda + (k0 + c)];
      if (k0 + KSTEP < K)  // gfx1250 global_prefetch_b8 of next A tile
        __builtin_prefetch(&A[(size_t)(m0 + r) * lda + (k0 + KSTEP + c)], 0, 1);
    }
    __syncthreads();

    // Per-tile B base pointers for this K-step; inner offsets are immediates.
    const float* Bpj[TN];
    #pragma unroll
    for (int j = 0; j < TN; ++j)
      Bpj[j] = B + (size_t)(k0 + 2 * half) * LDB + co[j] + l16;

    #pragma unroll
    for (int kk = 0; kk < KSTEP; kk += 4) {
      // A 16x4 fragment: lanes 0-15 hold K=kk,kk+1; lanes 16-31 K=kk+2,kk+3
      v2f a;
      a.x = As[l16 * KSTEP + kk + 2 * half + 0];
      a.y = As[l16 * KSTEP + kk + 2 * half + 1];
      #pragma unroll
      for (int j = 0; j < TN; ++j) {
        v2f b;                         // B 4x16 fragment, mirror layout
        b.x = Bpj[j][(size_t)(kk + 0) * LDB];
        b.y = Bpj[j][(size_t)(kk + 1) * LDB];
        acc[j] = __builtin_amdgcn_wmma_f32_16x16x4_f32(
            /*neg_a=*/false, a, /*neg_b=*/false, b,
            /*c_mod=*/(short)0, acc[j], /*reuse_a=*/false, /*reuse_b=*/false);
      }
    }
    __syncthreads();
  }

  // -------- epilogue (stores guarded; guards are wave-uniform scalars) -----
  float inv2sc = 0.f;
  if (EPI == 2) inv2sc = 2.0f / sqrtf(curvature(log_c));

  #pragma unroll
  for (int j = 0; j < TN; ++j) {
    if (n0 + 16 * j >= N) continue;      // wave-uniform
    const int col = n0 + 16 * j + l16;
    float p2 = 0.f, fp = 0.f, bv = 0.f;
    if (EPI == 2) { p2 = colP2[col]; fp = colFp[col]; }
    else if (bias) bv = bias[col];

    // C/D layout: VGPR r -> M = m0 + 8*half + r, N = col
    #pragma unroll
    for (int r = 0; r < 8; ++r) {
      const int row = m0 + half * 8 + r;
      float v = acc[j][r];
      if (EPI == 2) {
        const float e2 = rowE2[row];
        const float fe = rowFe[row];
        float diff = fmaxf(e2 + p2 - 2.0f * v, 1e-10f);
        float den  = fmaxf(fe * fp, 1e-6f);
        float arg  = fmaxf(1.0f + 2.0f * diff / den, 1.0f + 1e-6f);
        v = -inv2sc * acoshf(arg);
      } else {
        v += bv;
        if (EPI == 1) v = gelu_exact(v);
      }
      C[(size_t)row * ldc + col] = v;
    }
  }
}

// xs = LN(X + Y; g, b), rows of 128. One wave per row, 4 elems per lane.
__global__ __launch_bounds__(256)
void add_ln_kernel(const float* __restrict__ X, const float* __restrict__ Y,
                   const float* __restrict__ g, const float* __restrict__ b,
                   float* __restrict__ out, int nrows)
{
  const int row  = blockIdx.x * 8 + (threadIdx.x >> 5);
  const int lane = threadIdx.x & 31;
  if (row >= nrows) return;
  const float* xr = X + (size_t)row * 128;
  const float* yr = Y + (size_t)row * 128;
  float v[4];
  float s = 0.f;
  #pragma unroll
  for (int i = 0; i < 4; ++i) { v[i] = xr[lane + 32 * i] + yr[lane + 32 * i]; s += v[i]; }
  s = wave_sum(s);
  const float m = s * (1.0f / 128.0f);
  float q = 0.f;
  #pragma unroll
  for (int i = 0; i < 4; ++i) { const float d = v[i] - m; q += d * d; }
  q = wave_sum(q);
  const float inv = rsqrtf(q * (1.0f / 128.0f) + 1e-5f);
  #pragma unroll
  for (int i = 0; i < 4; ++i) {
    const int c = lane + 32 * i;
    out[(size_t)row * 128 + c] = (v[i] - m) * inv * g[c] + b[c];
  }
}

// Hyperbolic row embedding: emb, e2 = |emb|^2, fe = 1 - c*min(e2, 1-1e-5).
__global__ __launch_bounds__(256)
void head_kernel(const float* __restrict__ U, const float* __restrict__ log_c,
                 float* __restrict__ emb_out, float* __restrict__ e2_out,
                 float* __restrict__ fe_out, int nrows)
{
  const int row  = blockIdx.x * 8 + (threadIdx.x >> 5);
  const int lane = threadIdx.x & 31;
  if (row >= nrows) return;
  const float c  = curvature(log_c);
  const float sc = sqrtf(c);
  const float* t = U + (size_t)row * 128;
  float v[4];
  float s = 0.f;
  #pragma unroll
  for (int i = 0; i < 4; ++i) { v[i] = t[lane + 32 * i]; s += v[i] * v[i]; }
  s = wave_sum(s);
  const float vn    = fmaxf(sqrtf(s), 1e-10f);
  const float scale = tanhf(sc * vn * 0.5f) / (sc * vn);   // emb = scale * t
  const float en    = scale * vn;                           // ||emb||
  const float maxn  = (1.0f - 1e-5f) / sc;
  const float clip  = fminf(maxn / fmaxf(en, 1e-6f), 1.0f);
  const float tot   = scale * clip;
  #pragma unroll
  for (int i = 0; i < 4; ++i)
    emb_out[(size_t)row * 128 + lane + 32 * i] = tot * v[i];
  const float e2 = tot * tot * s;
  if (lane == 0) {
    e2_out[row] = e2;
    fe_out[row] = 1.0f - c * fminf(e2, 1.0f - 1e-5f);
  }
}

// Clip prototypes to the Poincare ball, write transposed PT (128 x C),
// plus p2 and fp = 1 - c*min(p2, 1-1e-5).
__global__ __launch_bounds__(256)
void proto_kernel(const float* __restrict__ P, const float* __restrict__ log_c,
                  float* __restrict__ PT, float* __restrict__ p2_out,
                  float* __restrict__ fp_out, int C)
{
  const int row  = blockIdx.x * 8 + (threadIdx.x >> 5);
  const int lane = threadIdx.x & 31;
  if (row >= C) return;
  const float c    = curvature(log_c);
  const float maxn = (1.0f - 1e-5f) / sqrtf(c);
  const float* pr  = P + (size_t)row * 128;
  float v[4];
  float s = 0.f;
  #pragma unroll
  for (int i = 0; i < 4; ++i) { v[i] = pr[lane + 32 * i]; s += v[i] * v[i]; }
  s = wave_sum(s);
  const float pn   = sqrtf(s);
  const float clip = fminf(maxn / fmaxf(pn, 1e-6f), 1.0f);
  #pragma unroll
  for (int i = 0; i < 4; ++i)
    PT[(size_t)(lane + 32 * i) * C + row] = clip * v[i];
  if (lane == 0) {
    const float p2 = clip * clip * s;
    p2_out[row] = p2;
    fp_out[row] = 1.0f - c * fminf(p2, 1.0f - 1e-5f);
  }
}

extern "C" void kernel_launch(void* const* d_in, const int* in_sizes, int n_in,
                              void* d_out, int out_size, void* d_ws, size_t ws_size,
                              hipStream_t stream)
{
  (void)in_sizes; (void)n_in; (void)out_size; (void)ws_size;
  const float* x      = (const float*)d_in[0];
  const float* W1     = (const float*)d_in[1];
  const float* b1     = (const float*)d_in[2];
  const float* W2     = (const float*)d_in[3];
  const float* b2     = (const float*)d_in[4];
  const float* in_w   = (const float*)d_in[5];
  const float* in_b   = (const float*)d_in[6];
  const float* out_w  = (const float*)d_in[7];
  const float* out_b  = (const float*)d_in[8];
  const float* fw1    = (const float*)d_in[9];
  const float* fb1    = (const float*)d_in[10];
  const float* fw2    = (const float*)d_in[11];
  const float* fb2    = (const float*)d_in[12];
  const float* g1     = (const float*)d_in[13];
  const float* bt1    = (const float*)d_in[14];
  const float* g2     = (const float*)d_in[15];
  const float* bt2    = (const float*)d_in[16];
  const float* log_c  = (const float*)d_in[17];
  const float* protos = (const float*)d_in[18];
  const float* euc_w  = (const float*)d_in[19];
  const float* euc_b  = (const float*)d_in[20];

  const int Bv = 4096, IN = 1024, D = 128, Cc = 10000;

  float* ws   = (float*)d_ws;
  float* h1   = ws;                         // 4096*256
  float* t    = h1   + (size_t)Bv * 256;    // 4096*128
  float* vbuf = t    + (size_t)Bv * D;      // 4096*128
  float* attn = vbuf + (size_t)Bv * D;      // 4096*128
  float* xs1  = attn + (size_t)Bv * D;      // 4096*128
  float* ffn1 = xs1  + (size_t)Bv * D;      // 4096*512
  float* ffn2 = ffn1 + (size_t)Bv * 512;    // 4096*128
  float* u    = ffn2 + (size_t)Bv * D;      // 4096*128
  float* e2a  = u    + (size_t)Bv * D;      // 4096
  float* fea  = e2a  + Bv;                  // 4096
  float* p2a  = fea  + Bv;                  // 10000
  float* fpa  = p2a  + Cc;                  // 10000
  float* PT   = fpa  + Cc;                  // 128*10000

  float* hyp = (float*)d_out;
  float* euc = hyp + (size_t)Bv * Cc;
  float* emb = euc + (size_t)Bv * Cc;

  const dim3 blk(256);
  auto grid = [](int M, int N, int TN) {
    const int colsPerBlock = 16 * TN * WPB;
    return dim3((N + colsPerBlock - 1) / colsPerBlock, M / 16);
  };
  const float* np = nullptr;

  // t = gelu(x@W1+b1) @ W2 + b2
  wmma_gemm_nn<1, 2, 256><<<grid(Bv, 256, 2), blk, 0, stream>>>(
      x, IN, W1, b1, h1, 256, Bv, 256, IN, np, np, np, np, np);
  wmma_gemm_nn<0, 1, 128><<<grid(Bv, D, 1), blk, 0, stream>>>(
      h1, 256, W2, b2, t, D, Bv, D, 256, np, np, np, np, np);
  // seq-len-1 attention: softmax over 1 element == 1 -> attn = (t@Wv+bv)@out_w+out_b
  wmma_gemm_nn<0, 1, 384><<<grid(Bv, D, 1), blk, 0, stream>>>(
      t, D, in_w + 2 * D, in_b + 2 * D, vbuf, D, Bv, D, D, np, np, np, np, np);
  wmma_gemm_nn<0, 1, 128><<<grid(Bv, D, 1), blk, 0, stream>>>(
      vbuf, D, out_w, out_b, attn, D, Bv, D, D, np, np, np, np, np);
  add_ln_kernel<<<Bv / 8, blk, 0, stream>>>(t, attn, g1, bt1, xs1, Bv);
  // FFN
  wmma_gemm_nn<1, 2, 512><<<grid(Bv, 512, 2), blk, 0, stream>>>(
      xs1, D, fw1, fb1, ffn1, 512, Bv, 512, D, np, np, np, np, np);
  wmma_gemm_nn<0, 1, 128><<<grid(Bv, D, 1), blk, 0, stream>>>(
      ffn1, 512, fw2, fb2, ffn2, D, Bv, D, 512, np, np, np, np, np);
  add_ln_kernel<<<Bv / 8, blk, 0, stream>>>(xs1, ffn2, g2, bt2, u, Bv);
  // Hyperbolic head prep
  head_kernel<<<Bv / 8, blk, 0, stream>>>(u, log_c, emb, e2a, fea, Bv);
  proto_kernel<<<(Cc + 7) / 8, blk, 0, stream>>>(protos, log_c, PT, p2a, fpa, Cc);
  // hyp_logits: acosh distance fused as GEMM epilogue (no 160MB temp)
  wmma_gemm_nn<2, 4, 10000><<<grid(Bv, Cc, 4), blk, 0, stream>>>(
      emb, D, PT, np, hyp, Cc, Bv, Cc, D, e2a, fea, p2a, fpa, log_c);
  // euc_logits = t @ euc_w + euc_b
  wmma_gemm_nn<0, 4, 10000><<<grid(Bv, Cc, 4), blk, 0, stream>>>(
      t, D, euc_w, euc_b, euc, Cc, Bv, Cc, D, np, np, np, np, np);
}